// MessagePassing_84482006712921
// MI455X (gfx1250) — compile-verified
//
#include <hip/hip_runtime.h>
#include <hip/hip_bf16.h>

typedef __attribute__((ext_vector_type(2))) float v2f;
typedef __attribute__((ext_vector_type(8))) float v8f;

#define N_NODES 100000
#define N_EDGES 640000
#define D_FEAT  128

// ---------------------------------------------------------------------------
// Kernel 1: zero the aggregation buffer A [N_NODES, 128] in workspace.
// N_NODES*128 = 12,800,000 floats = 3,200,000 float4 -> 12500 blocks x 256.
// ---------------------------------------------------------------------------
__global__ void __launch_bounds__(256) mp_zero_kernel(float4* __restrict__ A4) {
    const int i = blockIdx.x * 256 + threadIdx.x;
    A4[i] = make_float4(0.f, 0.f, 0.f, 0.f);
}

// ---------------------------------------------------------------------------
// Kernel 2: scatter-add raw source rows: A[tgt] += x[src].
// One wave (32 lanes) per edge; each lane handles a float4 chunk (32*16B=512B
// = one full 128-float row). x and A both fit in the 192MB L2, so the random
// gather + f32 atomics stay on-die.
// threads = E*32 = 20,480,000 -> 80000 blocks x 256.
// ---------------------------------------------------------------------------
__global__ void __launch_bounds__(256) mp_scatter_kernel(
        const float* __restrict__ x,
        const int*   __restrict__ ei,
        float*       __restrict__ A) {
    const int tid   = blockIdx.x * 256 + threadIdx.x;
    const int e     = tid >> 5;        // edge id
    const int chunk = tid & 31;        // which float4 of the 128-float row
    const int src = ei[e];
    const int tgt = ei[N_EDGES + e];
    const float4 v = *(const float4*)(x + (size_t)src * D_FEAT + chunk * 4);
    float* dst = A + (size_t)tgt * D_FEAT + chunk * 4;
    atomicAdd(dst + 0, v.x);
    atomicAdd(dst + 1, v.y);
    atomicAdd(dst + 2, v.z);
    atomicAdd(dst + 3, v.w);
}

// ---------------------------------------------------------------------------
// Kernel 3: out = A @ W + x @ W_self via V_WMMA_F32_16X16X4_F32 (full fp32).
// Block = 256 threads = 8 waves; block covers 16 rows x 128 cols, wave w
// owns the 16x16 tile at columns [16w, 16w+16).
//
// f32 16x16x4 fragment layout (ISA 7.12.2), lane = 16*h + l (h=lane/16):
//   A (16x4,  2 VGPRs): a[v] = Amat[row=l][k = 2h + v]
//   B (4x16,  2 VGPRs): b[v] = Bmat[k = 2h + v][col=l]
//   C/D (16x16, 8 VGPRs): c[v] = Cmat[row = v + 8h][col=l]
// Two independent accumulators (A@W and x@W_self) for WMMA ILP; summed at end.
// ---------------------------------------------------------------------------
__global__ void __launch_bounds__(256) mp_gemm_kernel(
        const float* __restrict__ A,
        const float* __restrict__ x,
        const float* __restrict__ W,
        const float* __restrict__ Ws,
        float*       __restrict__ out) {
    const int lane = threadIdx.x & 31;
    const int wave = threadIdx.x >> 5;
    const int l = lane & 15;
    const int h = lane >> 4;

    const int row0 = blockIdx.x * 16;
    const int row  = row0 + l;          // A-fragment row for this lane
    const int n    = wave * 16 + l;     // output column for this lane

    const float* __restrict__ arA = A + (size_t)row * D_FEAT;
    const float* __restrict__ arX = x + (size_t)row * D_FEAT;

    v8f c0 = {};   // accumulates A @ W
    v8f c1 = {};   // accumulates x @ W_self

    #pragma unroll
    for (int kt = 0; kt < 32; ++kt) {
        const int k = kt * 4 + 2 * h;   // this lane's K base within the K=4 step
        v2f aA = *(const v2f*)(arA + k);
        v2f aX = *(const v2f*)(arX + k);
        v2f bW, bS;
        bW[0] = W [(size_t)k * D_FEAT + n];
        bW[1] = W [(size_t)(k + 1) * D_FEAT + n];
        bS[0] = Ws[(size_t)k * D_FEAT + n];
        bS[1] = Ws[(size_t)(k + 1) * D_FEAT + n];
        c0 = __builtin_amdgcn_wmma_f32_16x16x4_f32(
                false, aA, false, bW, (short)0, c0, false, false);
        c1 = __builtin_amdgcn_wmma_f32_16x16x4_f32(
                false, aX, false, bS, (short)0, c1, false, false);
    }

    #pragma unroll
    for (int v = 0; v < 8; ++v) {
        out[(size_t)(row0 + v + 8 * h) * D_FEAT + n] = c0[v] + c1[v];
    }
}

// ---------------------------------------------------------------------------
extern "C" void kernel_launch(void* const* d_in, const int* in_sizes, int n_in,
                              void* d_out, int out_size, void* d_ws, size_t ws_size,
                              hipStream_t stream) {
    const float* x  = (const float*)d_in[0];
    const int*   ei = (const int*)  d_in[1];
    const float* W  = (const float*)d_in[2];
    const float* Ws = (const float*)d_in[3];
    float* out = (float*)d_out;
    float* Aagg = (float*)d_ws;   // N_NODES * D_FEAT floats = 51.2 MB

    // 1) zero aggregation buffer
    mp_zero_kernel<<<(N_NODES * D_FEAT) / (256 * 4), 256, 0, stream>>>((float4*)Aagg);

    // 2) scatter: A[tgt] += x[src]  (one wave per edge)
    mp_scatter_kernel<<<(N_EDGES * 32) / 256, 256, 0, stream>>>(x, ei, Aagg);

    // 3) out = A @ W + x @ W_self  (WMMA fp32)
    mp_gemm_kernel<<<N_NODES / 16, 256, 0, stream>>>(Aagg, x, W, Ws, out);
}